// FoutNet_54760833024353
// MI455X (gfx1250) — compile-verified
//
#include <hip/hip_runtime.h>
#include <hip/hip_bf16.h>

typedef __attribute__((ext_vector_type(2))) float v2f;
typedef __attribute__((ext_vector_type(8))) float v8f;

#define N_NODES   100000
#define F_IN      64
#define N2_NODES  25000
#define N3_NODES  5000
#define E_EDGES   (32 * N_NODES)
#define E2_EDGES  (16 * N2_NODES)

// ---------------------------------------------------------------------------
// Layer 1: alpha = x @ wc (100000x64 @ 64x16), beta = x @ wn, fp32 WMMA.
// One 16-row tile per wave; K=64 in steps of 4 via V_WMMA_F32_16X16X4_F32.
// A 16x4 layout: lane L holds row (L&15), VGPR0/1 = K = kbase+(L>>4)*2 (+1).
// B 4x16 layout: lane L holds col (L&15), VGPR0/1 = K = kbase+(L>>4)*2 (+1).
// D 16x16 layout: VGPR r, lane L -> row r+(L>>4)*8, col (L&15).
// ---------------------------------------------------------------------------
__global__ void __launch_bounds__(128)
l1_matmul_kernel(const float* __restrict__ x, const float* __restrict__ wc,
                 const float* __restrict__ wn, float* __restrict__ alpha,
                 float* __restrict__ beta) {
  const int wave = threadIdx.x >> 5;
  const int lane = threadIdx.x & 31;
  const int tile = blockIdx.x * 4 + wave;
  const int numTiles = N_NODES / 16;           // 6250, exact
  if (tile >= numTiles) return;                // whole-wave uniform exit

  const int row   = tile * 16 + (lane & 15);
  const int khalf = (lane >> 4) * 2;
  const int col   = lane & 15;

  v8f acc_a = {};
  v8f acc_b = {};
  for (int kk = 0; kk < F_IN; kk += 4) {
    const int ka = kk + khalf;
    v2f a;  a.x  = x[row * F_IN + ka];       a.y  = x[row * F_IN + ka + 1];
    v2f bc; bc.x = wc[ka * 16 + col];        bc.y = wc[(ka + 1) * 16 + col];
    v2f bn; bn.x = wn[ka * 16 + col];        bn.y = wn[(ka + 1) * 16 + col];
    acc_a = __builtin_amdgcn_wmma_f32_16x16x4_f32(false, a, false, bc,
                                                  (short)0, acc_a, false, false);
    acc_b = __builtin_amdgcn_wmma_f32_16x16x4_f32(false, a, false, bn,
                                                  (short)0, acc_b, false, false);
  }
  #pragma unroll
  for (int r = 0; r < 8; ++r) {
    const int m = tile * 16 + r + (lane >> 4) * 8;
    alpha[m * 16 + col] = acc_a[r];
    beta [m * 16 + col] = acc_b[r];
  }
}

// ---------------------------------------------------------------------------
// Layer 2: alpha2 = h @ wc2 (25000x16 @ 16x32), beta2 = h @ wn2.
// Two 16-wide column tiles; K=16 in steps of 4. Tail rows: clamped loads,
// guarded stores (EXEC remains all-1s around the WMMA ops).
// ---------------------------------------------------------------------------
__global__ void __launch_bounds__(128)
l2_matmul_kernel(const float* __restrict__ h, const float* __restrict__ wc,
                 const float* __restrict__ wn, float* __restrict__ alpha,
                 float* __restrict__ beta) {
  const int wave = threadIdx.x >> 5;
  const int lane = threadIdx.x & 31;
  const int tile = blockIdx.x * 4 + wave;
  const int numTiles = (N2_NODES + 15) / 16;   // 1563
  if (tile >= numTiles) return;

  int row = tile * 16 + (lane & 15);
  const int rowc  = row < N2_NODES ? row : (N2_NODES - 1);
  const int khalf = (lane >> 4) * 2;
  const int col   = lane & 15;

  v8f acc[4] = {v8f{}, v8f{}, v8f{}, v8f{}};   // [wc ct0, wc ct1, wn ct0, wn ct1]
  for (int kk = 0; kk < 16; kk += 4) {
    const int ka = kk + khalf;
    v2f a; a.x = h[rowc * 16 + ka]; a.y = h[rowc * 16 + ka + 1];
    #pragma unroll
    for (int ct = 0; ct < 2; ++ct) {
      const int c = ct * 16 + col;
      v2f bc; bc.x = wc[ka * 32 + c]; bc.y = wc[(ka + 1) * 32 + c];
      v2f bn; bn.x = wn[ka * 32 + c]; bn.y = wn[(ka + 1) * 32 + c];
      acc[ct]     = __builtin_amdgcn_wmma_f32_16x16x4_f32(false, a, false, bc,
                      (short)0, acc[ct], false, false);
      acc[2 + ct] = __builtin_amdgcn_wmma_f32_16x16x4_f32(false, a, false, bn,
                      (short)0, acc[2 + ct], false, false);
    }
  }
  #pragma unroll
  for (int r = 0; r < 8; ++r) {
    const int m = tile * 16 + r + (lane >> 4) * 8;
    if (m < N2_NODES) {
      alpha[m * 32 + col]      = acc[0][r];
      alpha[m * 32 + 16 + col] = acc[1][r];
      beta [m * 32 + col]      = acc[2][r];
      beta [m * 32 + 16 + col] = acc[3][r];
    }
  }
}

// One thread per (edge, feature): sums[src] += beta[dst]; cnt[src] += 1.
__global__ void edge_agg_kernel(const int* __restrict__ src,
                                const int* __restrict__ dst,
                                const float* __restrict__ beta,
                                float* __restrict__ sums,
                                float* __restrict__ cnt, int E, int F) {
  const int tid = blockIdx.x * blockDim.x + threadIdx.x;
  const int total = E * F;
  if (tid >= total) return;
  const int e = tid / F;
  const int f = tid - e * F;
  const int s = src[e];
  const int d = dst[e];
  atomicAdd(&sums[s * F + f], beta[d * F + f]);
  if (f == 0) atomicAdd(&cnt[s], 1.0f);
}

// h = relu(alpha + sums/cnt + bias), in place over alpha.
__global__ void combine_relu_kernel(float* __restrict__ alpha,
                                    const float* __restrict__ sums,
                                    const float* __restrict__ cnt,
                                    const float* __restrict__ bias,
                                    int n, int F) {
  const int tid = blockIdx.x * blockDim.x + threadIdx.x;
  if (tid >= n * F) return;
  const int i = tid / F;
  const int f = tid - i * F;
  const float v = alpha[tid] + sums[tid] / cnt[i] + bias[f];
  alpha[tid] = v > 0.0f ? v : 0.0f;
}

// Cluster mean pooling: accumulate.
__global__ void cluster_add_kernel(const float* __restrict__ h,
                                   const int* __restrict__ cluster,
                                   float* __restrict__ psum,
                                   float* __restrict__ pcnt, int n, int F) {
  const int tid = blockIdx.x * blockDim.x + threadIdx.x;
  if (tid >= n * F) return;
  const int i = tid / F;
  const int f = tid - i * F;
  const int c = cluster[i];
  atomicAdd(&psum[c * F + f], h[tid]);
  if (f == 0) atomicAdd(&pcnt[c], 1.0f);
}

__global__ void cluster_div_kernel(float* __restrict__ psum,
                                   const float* __restrict__ pcnt, int n, int F) {
  const int tid = blockIdx.x * blockDim.x + threadIdx.x;
  if (tid >= n * F) return;
  psum[tid] /= pcnt[tid / F];
}

// segment_max over post-ReLU (>=0) values via uint-bitpattern atomicMax.
__global__ void cluster_max_kernel(const float* __restrict__ h,
                                   const int* __restrict__ cluster,
                                   unsigned int* __restrict__ mx, int n, int F) {
  const int tid = blockIdx.x * blockDim.x + threadIdx.x;
  if (tid >= n * F) return;
  const int i = tid / F;
  const int f = tid - i * F;
  const int c = cluster[i];
  atomicMax(&mx[c * F + f], __float_as_uint(h[tid]));
}

// Head: g = mean(mx, axis=0) [5000x32 -> 32]; relu(g@fc1_w+fc1_b); @fc2_w+fc2_b.
__global__ void __launch_bounds__(256)
head_kernel(const float* __restrict__ mx, const float* __restrict__ fc1_w,
            const float* __restrict__ fc1_b, const float* __restrict__ fc2_w,
            const float* __restrict__ fc2_b, float* __restrict__ out) {
  __shared__ float part[256];
  __shared__ float g[32];
  __shared__ float z[64];
  const int t = threadIdx.x;
  const int f = t & 31;
  const int grp = t >> 5;                      // 8 row-groups
  float s = 0.0f;
  for (int r = grp; r < N3_NODES; r += 8) s += mx[r * 32 + f];
  part[t] = s;
  __syncthreads();
  if (t < 32) {
    float acc = 0.0f;
    #pragma unroll
    for (int gg = 0; gg < 8; ++gg) acc += part[gg * 32 + t];
    g[t] = acc / (float)N3_NODES;
  }
  __syncthreads();
  if (t < 64) {
    float acc = fc1_b[t];
    #pragma unroll
    for (int i = 0; i < 32; ++i) acc += g[i] * fc1_w[i * 64 + t];
    z[t] = acc > 0.0f ? acc : 0.0f;
  }
  __syncthreads();
  if (t == 0) {
    float acc = fc2_b[0];
    #pragma unroll
    for (int o = 0; o < 64; ++o) acc += z[o] * fc2_w[o];
    out[0] = acc;
  }
}

extern "C" void kernel_launch(void* const* d_in, const int* in_sizes, int n_in,
                              void* d_out, int out_size, void* d_ws, size_t ws_size,
                              hipStream_t stream) {
  const float* x     = (const float*)d_in[0];
  const float* wc1   = (const float*)d_in[1];
  const float* wn1   = (const float*)d_in[2];
  const float* b1    = (const float*)d_in[3];
  const float* wc2   = (const float*)d_in[4];
  const float* wn2   = (const float*)d_in[5];
  const float* b2    = (const float*)d_in[6];
  const float* fc1_w = (const float*)d_in[7];
  const float* fc1_b = (const float*)d_in[8];
  const float* fc2_w = (const float*)d_in[9];
  const float* fc2_b = (const float*)d_in[10];
  const int* ei1     = (const int*)d_in[11];   // [2, E]
  const int* ei2     = (const int*)d_in[12];   // [2, E2]
  const int* cl0     = (const int*)d_in[13];   // [N]
  const int* cl1     = (const int*)d_in[14];   // [N2]
  float* out = (float*)d_out;

  // ---- workspace layout (floats) ----
  float* ws = (float*)d_ws;
  size_t o = 0;
  float* alpha1 = ws + o; o += (size_t)N_NODES * 16;      // also h1 (in place)
  float* beta1  = ws + o; o += (size_t)N_NODES * 16;
  float* sums1  = ws + o; o += (size_t)N_NODES * 16;      // zeroed
  float* cnt1   = ws + o; o += (size_t)N_NODES;           // zeroed (contig w/ sums1)
  float* psum   = ws + o; o += (size_t)N2_NODES * 16;     // zeroed; also h2 in place
  float* pcnt   = ws + o; o += (size_t)N2_NODES;          // zeroed (contig w/ psum)
  float* alpha2 = ws + o; o += (size_t)N2_NODES * 32;     // also h3 (in place)
  float* beta2  = ws + o; o += (size_t)N2_NODES * 32;
  float* sums2  = ws + o; o += (size_t)N2_NODES * 32;     // zeroed
  float* cnt2   = ws + o; o += (size_t)N2_NODES;          // zeroed (contig w/ sums2)
  float* mx     = ws + o; o += (size_t)N3_NODES * 32;     // zeroed

  // Zero atomic accumulators (graph-capture-legal async memsets).
  hipMemsetAsync(sums1, 0, ((size_t)N_NODES * 16 + N_NODES) * sizeof(float), stream);
  hipMemsetAsync(psum,  0, ((size_t)N2_NODES * 16 + N2_NODES) * sizeof(float), stream);
  hipMemsetAsync(sums2, 0, ((size_t)N2_NODES * 32 + N2_NODES) * sizeof(float), stream);
  hipMemsetAsync(mx,    0, (size_t)N3_NODES * 32 * sizeof(float), stream);

  // Layer 1 matmuls (fp32 WMMA).
  {
    const int numTiles = N_NODES / 16;                    // 6250
    const int grid = (numTiles + 3) / 4;
    l1_matmul_kernel<<<grid, 128, 0, stream>>>(x, wc1, wn1, alpha1, beta1);
  }
  // Edge aggregation layer 1.
  {
    const int total = E_EDGES * 16;
    edge_agg_kernel<<<(total + 255) / 256, 256, 0, stream>>>(
        ei1, ei1 + E_EDGES, beta1, sums1, cnt1, E_EDGES, 16);
  }
  // h1 = relu(alpha1 + sums1/cnt1 + b1), in place.
  {
    const int total = N_NODES * 16;
    combine_relu_kernel<<<(total + 255) / 256, 256, 0, stream>>>(
        alpha1, sums1, cnt1, b1, N_NODES, 16);
  }
  // Community mean pooling -> h2 (= psum in place).
  {
    const int total = N_NODES * 16;
    cluster_add_kernel<<<(total + 255) / 256, 256, 0, stream>>>(
        alpha1, cl0, psum, pcnt, N_NODES, 16);
    const int total2 = N2_NODES * 16;
    cluster_div_kernel<<<(total2 + 255) / 256, 256, 0, stream>>>(
        psum, pcnt, N2_NODES, 16);
  }
  // Layer 2 matmuls (fp32 WMMA).
  {
    const int numTiles = (N2_NODES + 15) / 16;            // 1563
    const int grid = (numTiles + 3) / 4;
    l2_matmul_kernel<<<grid, 128, 0, stream>>>(psum, wc2, wn2, alpha2, beta2);
  }
  // Edge aggregation layer 2.
  {
    const int total = E2_EDGES * 32;
    edge_agg_kernel<<<(total + 255) / 256, 256, 0, stream>>>(
        ei2, ei2 + E2_EDGES, beta2, sums2, cnt2, E2_EDGES, 32);
  }
  // h3 = relu(alpha2 + sums2/cnt2 + b2), in place.
  {
    const int total = N2_NODES * 32;
    combine_relu_kernel<<<(total + 255) / 256, 256, 0, stream>>>(
        alpha2, sums2, cnt2, b2, N2_NODES, 32);
  }
  // segment_max -> mx (post-ReLU values >= 0, so uint atomicMax is exact).
  {
    const int total = N2_NODES * 32;
    cluster_max_kernel<<<(total + 255) / 256, 256, 0, stream>>>(
        alpha2, cl1, (unsigned int*)mx, N2_NODES, 32);
  }
  // Head: mean over N3 rows, fc1+relu, fc2 -> out[0].
  head_kernel<<<1, 256, 0, stream>>>(mx, fc1_w, fc1_b, fc2_w, fc2_b, out);
}